// GridSpinConv_60636348285006
// MI455X (gfx1250) — compile-verified
//
#include <hip/hip_runtime.h>
#include <hip/hip_bf16.h>

// Problem constants (match reference).
constexpr int kV = 1500;
constexpr int kE = 20000;
constexpr int kC = 24;
constexpr int kM = 32;    // message channels
constexpr int kD = 128;   // output channels
constexpr int kG = 128;   // grid points = PHI*THETA
constexpr int kT = 16;    // THETA (conv kernel length)
constexpr int kK = 256;   // M*PHI = GEMM K dim
constexpr float kTau = 10.0f;

typedef __attribute__((ext_vector_type(2))) float v2f;
typedef __attribute__((ext_vector_type(8))) float v8f;

// ---------------------------------------------------------------------------
// Prep: W'[d,i] = (1/16) * sum_h conv_w[d,i,h]  (conv + adaptive-avg-pool
// collapse exactly to this linear map for circular padding), and the
// spherical grid (3 x 128) used by the exp-sampling step.
// ---------------------------------------------------------------------------
__global__ __launch_bounds__(256) void gsc_prep(const float* __restrict__ conv_w,
                                                float* __restrict__ Wp,
                                                float* __restrict__ grid) {
  const int tid = blockIdx.x * blockDim.x + threadIdx.x;
  if (tid < kD * kK) {
    const float* src = conv_w + (size_t)tid * kT;
    float s = 0.0f;
#pragma unroll
    for (int h = 0; h < kT; ++h) s += src[h];
    Wp[tid] = s * (1.0f / 16.0f);
  }
  if (tid < kG) {
    const int p = tid >> 4;       // phi index (8)
    const int t = tid & 15;       // theta index (16)
    const float PI = 3.14159265358979323846f;
    const float phi0 = PI / 16.0f, phi1 = PI - PI / 16.0f;
    const float th0 = PI / 16.0f, th1 = 2.0f * PI - PI / 16.0f;
    const float ph = phi0 + (float)p * ((phi1 - phi0) / 7.0f);
    const float th = th0 + (float)t * ((th1 - th0) / 15.0f);
    const float r = sinf(ph);
    grid[tid]           = r * cosf(th);   // gx
    grid[kG + tid]      = r * sinf(th);   // gy
    grid[2 * kG + tid]  = cosf(ph);       // gz
  }
}

// ---------------------------------------------------------------------------
// Main fused kernel: 16 edges per block, 8 waves of 32.
// Phase 1: build S_tile (16 x 256) in LDS. Phase 2: out_tile = S_tile @ W'^T
// via V_WMMA_F32_16X16X4_F32 (full fp32 matrix path).
// ---------------------------------------------------------------------------
__global__ __launch_bounds__(256) void gsc_main(const int* __restrict__ tei,
                                                const int* __restrict__ esi,
                                                const float* __restrict__ eu,
                                                const float* __restrict__ em,
                                                const float* __restrict__ Wp,
                                                const float* __restrict__ grid,
                                                float* __restrict__ out) {
  __shared__ float sS[16 * kK];  // 16 KB: S rows for the 16 edges of this block

  const int tid  = threadIdx.x;
  const int wave = tid >> 5;
  const int lane = tid & 31;
  const int blk  = blockIdx.x;

  // Warm the L2-resident weight slice this wave will consume in phase 2.
  __builtin_prefetch(Wp + (size_t)(wave * 16) * kK, 0, 1);

  // Each lane owns 4 consecutive grid points (one quarter of a theta ring).
  const int g0 = lane * 4;
  float gx[4], gy[4], gz[4];
#pragma unroll
  for (int j = 0; j < 4; ++j) {
    gx[j] = grid[g0 + j];
    gy[j] = grid[kG + g0 + j];
    gz[j] = grid[2 * kG + g0 + j];
  }

  // ----------------- Phase 1: S rows (2 edges per wave) ------------------
#pragma unroll
  for (int it = 0; it < 2; ++it) {
    const int eloc = wave + it * 8;
    const int e = blk * 16 + eloc;

    // Rodrigues rotation Q with Q @ u = e_z; local = Q @ a.
    const float ux = eu[e * 3 + 0], uy = eu[e * 3 + 1], uz = eu[e * 3 + 2];
    const float s  = sqrtf(ux * ux + uy * uy);
    const float cz = uz;
    const bool  safe = s > 1e-7f;
    const float inv = safe ? 1.0f / fmaxf(s, 1e-7f) : 0.0f;
    const float kx = safe ? uy * inv : 1.0f;
    const float ky = safe ? -ux * inv : 0.0f;
    const float oc = 1.0f - cz;
    const float Q00 = cz + oc * kx * kx, Q01 = oc * kx * ky, Q02 = s * ky;
    const float Q10 = Q01, Q11 = cz + oc * ky * ky, Q12 = -s * kx;
    const float Q20 = -s * ky, Q21 = s * kx, Q22 = cz;

    const int v = esi[e];

    float acc[8];
#pragma unroll
    for (int p = 0; p < 8; ++p) acc[p] = 0.0f;

    for (int c = 0; c < kC; ++c) {
      const int te = tei[v * kC + c];               // double indirection
      const float msg = em[te * kM + lane];         // lane owns channel m=lane
      const float a0 = eu[te * 3 + 0], a1 = eu[te * 3 + 1], a2 = eu[te * 3 + 2];
      const float l0 = Q00 * a0 + Q01 * a1 + Q02 * a2;
      const float l1 = Q10 * a0 + Q11 * a1 + Q12 * a2;
      const float l2 = Q20 * a0 + Q21 * a1 + Q22 * a2;

      float part = 0.0f;
#pragma unroll
      for (int j = 0; j < 4; ++j) {
        const float dot = l0 * gx[j] + l1 * gy[j] + l2 * gz[j];
        part += __expf(kTau * (dot - 1.0f));
      }
      // Reduce the 4 lanes of each phi band -> Q[c, p] in every lane of group.
      part += __shfl_xor(part, 1, 32);
      part += __shfl_xor(part, 2, 32);
      // Outer-product accumulate: S[m*8+p] += msg[c,m] * Q[c,p].
#pragma unroll
      for (int p = 0; p < 8; ++p) {
        const float qp = __shfl(part, p * 4, 32);
        acc[p] = fmaf(msg, qp, acc[p]);
      }
    }

    // S row layout: i = m*8 + p, lane m owns the contiguous 8-float slice.
    float4* row4 = reinterpret_cast<float4*>(sS + eloc * kK + lane * 8);
    row4[0] = make_float4(acc[0], acc[1], acc[2], acc[3]);
    row4[1] = make_float4(acc[4], acc[5], acc[6], acc[7]);
  }

  __syncthreads();

  // --------- Phase 2: out_tile(16x16 per wave) via WMMA f32 16x16x4 ---------
  // A (16x4 f32): lanes 0-15 -> M=lane, K={k0,k0+1}; lanes 16-31 -> K={k0+2,k0+3}
  // B (4x16 f32): lanes 0-15 -> N=lane, K={k0,k0+1}; lanes 16-31 -> K={k0+2,k0+3}
  const int row = lane & 15;          // M for A, N for B
  const int hi  = lane >> 4;
  const float* sSrow = sS + row * kK;                       // A row (edge)
  const float* sWrow = Wp + (size_t)(wave * 16 + row) * kK; // B col = W' row d

  v8f cacc = {};
#pragma unroll 8
  for (int kk = 0; kk < kK / 4; ++kk) {
    const int kb = kk * 4 + hi * 2;
    v2f a, b;
    a.x = sSrow[kb];
    a.y = sSrow[kb + 1];
    b.x = sWrow[kb];
    b.y = sWrow[kb + 1];
    cacc = __builtin_amdgcn_wmma_f32_16x16x4_f32(
        /*neg_a=*/false, a, /*neg_b=*/false, b,
        /*c_mod=*/(short)0, cacc, /*reuse_a=*/false, /*reuse_b=*/false);
  }

  // D layout: VGPR r, lanes 0-15 -> M=r, lanes 16-31 -> M=r+8; N = lane&15.
  const int ebase = blk * 16;
#pragma unroll
  for (int r = 0; r < 8; ++r) {
    const int Mrow = r + hi * 8;
    out[(size_t)(ebase + Mrow) * kD + wave * 16 + row] = cacc[r];
  }
}

// ---------------------------------------------------------------------------
extern "C" void kernel_launch(void* const* d_in, const int* in_sizes, int n_in,
                              void* d_out, int out_size, void* d_ws, size_t ws_size,
                              hipStream_t stream) {
  const int*   tei = (const int*)d_in[0];    // (V, C) int32
  const int*   esi = (const int*)d_in[1];    // (E,)   int32
  const float* eu  = (const float*)d_in[2];  // (E, 3) f32
  const float* em  = (const float*)d_in[3];  // (E, M) f32
  const float* cw  = (const float*)d_in[4];  // (D, M*PHI, THETA) f32
  float* out = (float*)d_out;                // (E, D) f32

  float* Wp   = (float*)d_ws;                // 128*256 f32
  float* grid = Wp + kD * kK;                // 3*128 f32

  gsc_prep<<<(kD * kK + 255) / 256, 256, 0, stream>>>(cw, Wp, grid);
  gsc_main<<<kE / 16, 256, 0, stream>>>(tei, esi, eu, em, Wp, grid, out);
}